// TransINTModelTestTransH_42520176230878
// MI455X (gfx1250) — compile-verified
//
#include <hip/hip_runtime.h>
#include <stdint.h>

// ---------------------------------------------------------------------------
// TransH scoring (memory-bound gather): one wave per triple-batch, CDNA5
// async gather-to-LDS double-buffered prefetch pipeline, wave32 reductions.
// Uses projection linearity: P(h)+P(r)-P(t) == P(h+r-t).
// Round 2: hoist all index loads out of the pipeline so the steady-state
// loop is pure {async-issue, s_wait_asynccnt, ds_read, VALU} with no
// loadcnt drains on the critical path.
// ---------------------------------------------------------------------------

#define WAVES_PER_BLOCK 8
#define BLOCK_THREADS   (WAVES_PER_BLOCK * 32)
#define TPW             4            // triples processed per wave
#define DDIM            128          // embedding dim
#define ROWS            7            // v, hp, tp, rp, hn, tn, rn
#define ROW_BYTES       (DDIM * 4)   // 512 B
#define BUF_FLOATS      (ROWS * DDIM)        // 896 floats / buffer
#define WAVE_FLOATS     (2 * BUF_FLOATS)     // double buffered
#define LDS_FLOATS      (WAVES_PER_BLOCK * WAVE_FLOATS)  // 14336 floats = 56 KB

// One wave gathers one 512B row: each of 32 lanes async-copies 16 bytes
// from global to its LDS slot. Tracked on ASYNCcnt (CDNA5 async path).
__device__ __forceinline__ void async_row_gather(unsigned lds_row_base,
                                                 const float* g_row, int lane) {
  unsigned dst = lds_row_base + (unsigned)(lane * 16);
  uint64_t src = (uint64_t)(uintptr_t)(g_row + lane * 4);
  asm volatile("global_load_async_to_lds_b128 %0, %1, off"
               :
               : "v"(dst), "v"(src)
               : "memory");
}

__device__ __forceinline__ void wait_async_le7() {
  asm volatile("s_wait_asynccnt 0x7" ::: "memory");
}
__device__ __forceinline__ void wait_async_le0() {
  asm volatile("s_wait_asynccnt 0x0" ::: "memory");
}

extern "C" __global__ void __launch_bounds__(BLOCK_THREADS)
transh_score_kernel(const int* __restrict__ pos_h, const int* __restrict__ pos_t,
                    const int* __restrict__ pos_r, const int* __restrict__ neg_h,
                    const int* __restrict__ neg_t, const int* __restrict__ neg_r,
                    const float* __restrict__ ent,  const float* __restrict__ vvrel,
                    const float* __restrict__ bases, float* __restrict__ out, int n) {
  __shared__ __align__(16) float lds[LDS_FLOATS];

  const int lane  = (int)(threadIdx.x & 31u);
  const int wslot = (int)(threadIdx.x >> 5);
  const int wave  = (int)blockIdx.x * WAVES_PER_BLOCK + wslot;
  const int i0    = wave * TPW;
  if (i0 >= n) return;                        // wave-uniform exit
  const int cnt = (n - i0 < TPW) ? (n - i0) : TPW;

  // ---- Hoisted index gather: 24 clamped (always-valid) loads, one wait. ----
  int idx_r[TPW], idx_ph[TPW], idx_pt[TPW], idx_nh[TPW], idx_nt[TPW], idx_nr[TPW];
#pragma unroll
  for (int j = 0; j < TPW; ++j) {
    const int ii = (i0 + j < n) ? (i0 + j) : (n - 1);  // branchless clamp
    idx_r[j]  = pos_r[ii];
    idx_ph[j] = pos_h[ii];
    idx_pt[j] = pos_t[ii];
    idx_nh[j] = neg_h[ii];
    idx_nt[j] = neg_t[ii];
    idx_nr[j] = neg_r[ii];
  }

  float* wbuf = &lds[wslot * WAVE_FLOATS];
  // Generic LDS pointer: low 32 bits are the LDS byte offset (aperture in
  // the high bits), matching the LDS address the async instruction takes from its VGPR.
  const unsigned lds_base = (unsigned)(uintptr_t)wbuf;

  auto prefetch = [&](int j, int buf) {
    const unsigned b = lds_base + (unsigned)(buf * (BUF_FLOATS * 4));
    async_row_gather(b + 0u * ROW_BYTES, bases + (size_t)idx_r[j]  * DDIM, lane);
    async_row_gather(b + 1u * ROW_BYTES, ent   + (size_t)idx_ph[j] * DDIM, lane);
    async_row_gather(b + 2u * ROW_BYTES, ent   + (size_t)idx_pt[j] * DDIM, lane);
    async_row_gather(b + 3u * ROW_BYTES, vvrel + (size_t)idx_r[j]  * DDIM, lane);
    async_row_gather(b + 4u * ROW_BYTES, ent   + (size_t)idx_nh[j] * DDIM, lane);
    async_row_gather(b + 5u * ROW_BYTES, ent   + (size_t)idx_nt[j] * DDIM, lane);
    async_row_gather(b + 6u * ROW_BYTES, vvrel + (size_t)idx_nr[j] * DDIM, lane);
  };

  prefetch(0, 0);

#pragma unroll
  for (int j = 0; j < TPW; ++j) {
    if (j >= cnt) break;                      // wave-uniform
    const int cur = j & 1;
    if (j + 1 < cnt) {
      prefetch(j + 1, cur ^ 1);   // issue next batch (7 async ops)
      wait_async_le7();           // previous batch of 7 now complete
    } else {
      wait_async_le0();           // drain
    }

    const float* b = wbuf + cur * BUF_FLOATS;
    const float4 v  = *(const float4*)(b + 0 * DDIM + lane * 4);
    const float4 hp = *(const float4*)(b + 1 * DDIM + lane * 4);
    const float4 tp = *(const float4*)(b + 2 * DDIM + lane * 4);
    const float4 rp = *(const float4*)(b + 3 * DDIM + lane * 4);
    const float4 hn = *(const float4*)(b + 4 * DDIM + lane * 4);
    const float4 tn = *(const float4*)(b + 5 * DDIM + lane * 4);
    const float4 rn = *(const float4*)(b + 6 * DDIM + lane * 4);

    // s = h + r - t (projection is linear, so project s once)
    const float spx = hp.x + rp.x - tp.x, spy = hp.y + rp.y - tp.y;
    const float spz = hp.z + rp.z - tp.z, spw = hp.w + rp.w - tp.w;
    const float snx = hn.x + rn.x - tn.x, sny = hn.y + rn.y - tn.y;
    const float snz = hn.z + rn.z - tn.z, snw = hn.w + rn.w - tn.w;

    float vv  = v.x * v.x + v.y * v.y + v.z * v.z + v.w * v.w;
    float vsp = v.x * spx + v.y * spy + v.z * spz + v.w * spw;
    float vsn = v.x * snx + v.y * sny + v.z * snz + v.w * snw;

#pragma unroll
    for (int m = 16; m >= 1; m >>= 1) {   // wave32 butterfly: all lanes get sum
      vv  += __shfl_xor(vv,  m, 32);
      vsp += __shfl_xor(vsp, m, 32);
      vsn += __shfl_xor(vsn, m, 32);
    }

    const float cp = vsp / vv;
    const float cn = vsn / vv;

    float p = fabsf(spx - cp * v.x) + fabsf(spy - cp * v.y) +
              fabsf(spz - cp * v.z) + fabsf(spw - cp * v.w);
    float q = fabsf(snx - cn * v.x) + fabsf(sny - cn * v.y) +
              fabsf(snz - cn * v.z) + fabsf(snw - cn * v.w);

#pragma unroll
    for (int m = 16; m >= 1; m >>= 1) {
      p += __shfl_xor(p, m, 32);
      q += __shfl_xor(q, m, 32);
    }

    if (lane == 0) {
      out[i0 + j]     = p;   // pos scores
      out[n + i0 + j] = q;   // neg scores (concatenated tuple output)
    }
  }
}

extern "C" void kernel_launch(void* const* d_in, const int* in_sizes, int n_in,
                              void* d_out, int out_size, void* d_ws, size_t ws_size,
                              hipStream_t stream) {
  const int*   pos_h = (const int*)d_in[0];
  const int*   pos_t = (const int*)d_in[1];
  const int*   pos_r = (const int*)d_in[2];
  const int*   neg_h = (const int*)d_in[3];
  const int*   neg_t = (const int*)d_in[4];
  const int*   neg_r = (const int*)d_in[5];
  const float* ent   = (const float*)d_in[6];
  const float* vvrel = (const float*)d_in[7];
  const float* bases = (const float*)d_in[8];
  float* out = (float*)d_out;

  const int n = in_sizes[0];
  const int triples_per_block = WAVES_PER_BLOCK * TPW;  // 32
  const int blocks = (n + triples_per_block - 1) / triples_per_block;

  hipLaunchKernelGGL(transh_score_kernel, dim3(blocks), dim3(BLOCK_THREADS), 0,
                     stream, pos_h, pos_t, pos_r, neg_h, neg_t, neg_r,
                     ent, vvrel, bases, out, n);
}